// RPQWeight_3255585210642
// MI455X (gfx1250) — compile-verified
//
#include <hip/hip_runtime.h>
#include <cstdint>

// Problem constants (match reference)
#define NUM_CODEBOOKS 32
#define CODEBOOK_DIM  128
#define NUM_VECTORS   16384
#define NUM_CENTROIDS 256

// Tunables
#define C_PER_BLOCK   512           // vectors (columns) handled per block
#define THREADS       256           // 8 wave32s
#define WAVES         (THREADS / 32)
#define CB_BYTES      (NUM_CENTROIDS * CODEBOOK_DIM * 4)   // 131072 = 128 KB
#define IDX_BYTES     (C_PER_BLOCK * 4)                    // 2 KB

// Native 16-byte vector type (valid operand for __builtin_nontemporal_store,
// unlike HIP's class-based float4).
typedef float v4f __attribute__((ext_vector_type(4)));

// out[c, h*128 + d] = codebooks[h, indices[h,c], d]
// Each block: one codebook h, a contiguous range of 512 c's.
// Stage codebook[h] (128 KB) + this block's 512 indices (2 KB) into LDS with
// CDNA5 async global->LDS DMA loads, then gather from LDS and stream 512B
// contiguous non-temporal b128 stores to the output.
__global__ __launch_bounds__(THREADS) void rpq_gather_kernel(
    const int*   __restrict__ indices,    // (32, 16384) i32
    const float* __restrict__ codebooks,  // (32, 256, 128) f32
    float*       __restrict__ out)        // (16384, 4096) f32
{
    extern __shared__ char smem[];        // [0,128K): codebook  [128K,130K): indices

    const int h    = blockIdx.x;                 // 0..31
    const int c0   = blockIdx.y * C_PER_BLOCK;   // base vector index
    const int lane = threadIdx.x & 31;
    const int wave = threadIdx.x >> 5;

    // Low 32 bits of a generic shared pointer == LDS byte offset (aperture scheme).
    const uint32_t lds_base = (uint32_t)(uintptr_t)smem;

    const char* cb_g  = (const char*)(codebooks + (size_t)h * NUM_CENTROIDS * CODEBOOK_DIM);
    const char* idx_g = (const char*)(indices + (size_t)h * NUM_VECTORS + c0);

    // ---- Async stage: codebook 128 KB = 32 iters x (256 thr x 16 B) ----
    #pragma unroll
    for (int i = 0; i < CB_BYTES / (THREADS * 16); ++i) {
        const uint32_t off      = (uint32_t)(i * THREADS * 16 + threadIdx.x * 16);
        const uint32_t lds_addr = lds_base + off;
        const uint64_t gaddr    = (uint64_t)(uintptr_t)(cb_g + off);
        asm volatile("global_load_async_to_lds_b128 %0, %1, off"
                     :: "v"(lds_addr), "v"(gaddr) : "memory");
    }
    // ---- Async stage: 512 indices = 2 KB = 128 thr x 16 B ----
    if (threadIdx.x < IDX_BYTES / 16) {
        const uint32_t lds_addr = lds_base + CB_BYTES + threadIdx.x * 16;
        const uint64_t gaddr    = (uint64_t)(uintptr_t)(idx_g + threadIdx.x * 16);
        asm volatile("global_load_async_to_lds_b128 %0, %1, off"
                     :: "v"(lds_addr), "v"(gaddr) : "memory");
    }
    // Wait for this wave's async DMAs, then barrier so all waves see full LDS.
    asm volatile("s_wait_asynccnt 0x0" ::: "memory");
    __syncthreads();

    const v4f* lds_cb4 = (const v4f*)smem;                     // 32 v4f per centroid row
    const int* lds_idx = (const int*)(smem + CB_BYTES);

    v4f* out4 = (v4f*)out;
    // Output row c = 4096 floats = 1024 v4f; this h occupies v4f [h*32, h*32+32).
    const int per_wave = C_PER_BLOCK / WAVES;                  // 64
    const int cc0      = wave * per_wave;

    #pragma unroll 4
    for (int i = 0; i < per_wave; ++i) {
        const int cc  = cc0 + i;
        const int idx = lds_idx[cc];                           // broadcast LDS read
        const v4f v   = lds_cb4[idx * 32 + lane];              // conflict-free b128 LDS read
        // 512 contiguous bytes per wave-instruction; NT: don't pollute L2 with the 256MB stream
        __builtin_nontemporal_store(v, &out4[(size_t)(c0 + cc) * (NUM_CODEBOOKS * CODEBOOK_DIM / 4)
                                             + h * (CODEBOOK_DIM / 4) + lane]);
    }
}

extern "C" void kernel_launch(void* const* d_in, const int* in_sizes, int n_in,
                              void* d_out, int out_size, void* d_ws, size_t ws_size,
                              hipStream_t stream) {
    const int*   indices   = (const int*)d_in[0];    // (32, 16384) int32
    const float* codebooks = (const float*)d_in[1];  // (32, 256, 128) float32
    float*       out       = (float*)d_out;          // (16384, 4096) float32

    dim3 grid(NUM_CODEBOOKS, NUM_VECTORS / C_PER_BLOCK);  // (32, 32)
    dim3 block(THREADS);
    size_t shmem = CB_BYTES + IDX_BYTES;                  // 130 KB (<320 KB WGP LDS)
    hipLaunchKernelGGL(rpq_gather_kernel, grid, block, shmem, stream,
                       indices, codebooks, out);
}